// LSTM_BaselineAdvanced_50190987821348
// MI455X (gfx1250) — compile-verified
//
#include <hip/hip_runtime.h>
#include <hip/hip_bf16.h>
#include <math.h>

// ---------------------------------------------------------------------------
// Model dims (from reference)
// ---------------------------------------------------------------------------
#define T_N    2048
#define IN_D   78
#define OUT_D  11
#define H_D    256
#define E_D    128
#define Z_D    16
#define FH_D   512
#define NPM_D  16
#define FIN_D  542   // 2H + OUT + 1 + 2 + Z

// ---------------------------------------------------------------------------
// d_in index map (jax tree-flatten: top-level dict keys sorted; within LSTM
// param dicts the sort order is Whh < Wih < b).
//  0:x  1:y  2:measure_numbers
//  3:contract.W(128x523)     4:contract.b
//  5:fc1.W(256x78)  6:fc1.b  7:fc2.W  8:fc2.b  9:fc3.W 10:fc3.b
// 11:fc_out.W(11x512) 12:fc_out.b
// 13:final_att.W(256x256) 14:final_att.Wb 15:final_att.ctx(8x32)
// 16..18: lstm[0][fwd]  Whh(1024x256), Wih(1024x256),  b(1024)
// 19..21: lstm[0][bwd]
// 22..24: lstm[1][fwd]  Whh(1024x256), Wih(1024x512),  b(1024)
// 25..27: lstm[1][bwd]
// 28:mean.W(16x256) 29:mean.b
// 30:meas_att.W(256x256) 31:meas_att.Wb 32:meas_att.ctx(8x32)
// 33..35: note_enc[fwd] Whh(512x128), Wih(512x128), b(512)
// 36..38: note_enc[bwd]
// 39:out_lstm.Whh(2048x512) 40:out_lstm.Wih(2048x542) 41:out_lstm.b(2048)
// 42..44: perf_enc[0][fwd] Whh(512x128), Wih(512x256), b(512)
// 45..47: perf_enc[0][bwd]
// 48..50: perf_enc[1][fwd]
// 51..53: perf_enc[1][bwd]
// 54:var.W(16x256) 55:var.b
// ---------------------------------------------------------------------------

typedef __attribute__((ext_vector_type(2))) float v2f;
typedef __attribute__((ext_vector_type(8))) float v8f;

__device__ __forceinline__ float sigf(float x) { return 1.0f / (1.0f + __expf(-x)); }

// eps = jax.random.normal(jax.random.key(42), (16,)) — fixed, input-independent;
// values precomputed host-side once.
__device__ __constant__ float EPS16[16] = {
    -0.71668226f,  0.31629813f,  0.76456165f, -1.26142892f,
     0.54027724f, -0.09462691f, -1.52973449f,  0.03755537f,
     1.10028958f, -0.44521444f,  0.28916568f, -0.86713219f,
    -0.13755953f,  1.43771374f,  0.02342212f, -0.60593784f
};

// ---------------------------------------------------------------------------
// CDNA5 async global->LDS copy (ASYNCcnt) + wait. Flat LDS pointers truncate
// to the wave's LDS byte offset (ISA 10.2: low 32 address bits).
// ---------------------------------------------------------------------------
__device__ __forceinline__ void async_ld_f32(const float* gsrc, float* ldst)
{
    asm volatile("global_load_async_to_lds_b32 %0, %1, off"
                 :: "v"((unsigned int)(unsigned long long)(uintptr_t)ldst),
                    "v"((unsigned long long)(uintptr_t)gsrc)
                 : "memory");
}

__device__ __forceinline__ void wait_async0()
{
#if __has_builtin(__builtin_amdgcn_s_wait_asynccnt)
    __builtin_amdgcn_s_wait_asynccnt(0);
#else
    asm volatile("s_wait_asynccnt 0x0" ::: "memory");
#endif
}

// ---------------------------------------------------------------------------
// FP32 WMMA GEMM:  C[M x N] = act(A[M x K] @ W[N x K]^T + bias)
// Block = 4 waves. All 4 waves share one 16-col N tile; wave w owns the
// 16-row M tile m0 = (blockIdx.y*4 + w)*16. B chunk (16x32) is staged in LDS
// via async global->LDS loads; A streams straight from global (L2-resident).
// Launched shapes always have 16 | M and 16 | N, so only the K tail is masked
// (address-clamp + value-select: no EXEC branching, WMMA stays legal).
// ---------------------------------------------------------------------------
#define GEMM_KC 32

template<bool CHECKED>
__device__ __forceinline__ void stage_B(const float* __restrict__ W, float* Bs,
                                        int n0, int ldw, int k, int K, int tid)
{
    for (int e = tid; e < 16 * GEMM_KC; e += 128) {
        const int row = e >> 5, col = e & 31;
        float* dst = Bs + row * 33 + col;
        if (!CHECKED || (k + col) < K) {
            async_ld_f32(W + (size_t)(n0 + row) * ldw + (k + col), dst);
        } else {
            *dst = 0.f;
        }
    }
}

template<bool CHECKED>
__device__ __forceinline__ v8f compute_chunk(v8f acc, const float* __restrict__ Arow,
                                             const float* Bs, int k, int K, int lane)
{
    const int nn = lane & 15;
    const int ko = (lane >> 4) << 1;
    #pragma unroll
    for (int kk = 0; kk < GEMM_KC; kk += 4) {
        const int k0 = k + kk + ko;
        v2f a, b;
        if (CHECKED) {
            const int kc0 = (k0     < K) ? k0       : (K - 1);  // clamped address
            const int kc1 = (k0 + 1 < K) ? (k0 + 1) : (K - 1);
            const float a0 = Arow[kc0];   // unconditional loads
            const float a1 = Arow[kc1];
            a.x = (k0     < K) ? a0 : 0.f;  // value select, no EXEC change
            a.y = (k0 + 1 < K) ? a1 : 0.f;
        } else {
            a.x = Arow[k0];
            a.y = Arow[k0 + 1];
        }
        b.x = Bs[nn * 33 + kk + ko];
        b.y = Bs[nn * 33 + kk + ko + 1];
        acc = __builtin_amdgcn_wmma_f32_16x16x4_f32(
                  false, a, false, b, (short)0, acc, false, false);
    }
    return acc;
}

__global__ void gemm_wmma_f32(const float* __restrict__ A,
                              const float* __restrict__ W,
                              const float* __restrict__ bias,
                              float* __restrict__ C,
                              int M, int N, int K,
                              int lda, int ldw, int ldc, int act)
{
    __shared__ __align__(16) float Bs[16 * 33];
    const int lane = threadIdx.x;            // 0..31
    const int wave = threadIdx.y;            // 0..3
    const int tid  = wave * 32 + lane;
    const int n0   = blockIdx.x << 4;
    const int m0   = ((blockIdx.y << 2) + wave) << 4;

    const float* __restrict__ Arow = A + (size_t)(m0 + (lane & 15)) * lda;

    v8f acc = {0.f, 0.f, 0.f, 0.f, 0.f, 0.f, 0.f, 0.f};

    int k = 0;
    for (; k + GEMM_KC <= K; k += GEMM_KC) {
        __syncthreads();
        stage_B<false>(W, Bs, n0, ldw, k, K, tid);
        wait_async0();
        __syncthreads();
        acc = compute_chunk<false>(acc, Arow, Bs, k, K, lane);
    }
    if (k < K) {
        __syncthreads();
        stage_B<true>(W, Bs, n0, ldw, k, K, tid);
        wait_async0();
        __syncthreads();
        acc = compute_chunk<true>(acc, Arow, Bs, k, K, lane);
    }

    const int n = n0 + (lane & 15);
    const float bv = bias ? bias[n] : 0.f;
    #pragma unroll
    for (int v = 0; v < 8; ++v) {
        const int m = m0 + v + ((lane >> 4) << 3);
        float rr = acc[v] + bv;
        if (act == 1) rr = fmaxf(rr, 0.f);
        C[(size_t)m * ldc + n] = rr;
    }
}

// ---------------------------------------------------------------------------
// Bidirectional LSTM recurrence. Gx = x@Wih^T + b precomputed (WMMA GEMM).
// blockIdx.x: 0 = forward, 1 = backward. 1024 threads (32 waves fill a WGP).
// Per step: g[j] = Gx[t][j] + Whh[j,:]·h (float4, L2-resident weights).
// ---------------------------------------------------------------------------
__global__ void bilstm_seq(const float* __restrict__ Gx_f,
                           const float* __restrict__ Whh_f,
                           const float* __restrict__ Gx_b,
                           const float* __restrict__ Whh_b,
                           float* __restrict__ out,
                           int T, int Hh, int outStride)
{
    const int dir = blockIdx.x;
    const float* __restrict__ Gx  = dir ? Gx_b  : Gx_f;
    const float* __restrict__ Whh = dir ? Whh_b : Whh_f;
    const int G = 4 * Hh;
    const int H4 = Hh >> 2;

    __shared__ __align__(16) float h[256];
    __shared__ float c[256];
    __shared__ float g[1024];

    const int tid = threadIdx.x;
    const int nth = blockDim.x;

    for (int j = tid; j < Hh; j += nth) { h[j] = 0.f; c[j] = 0.f; }
    __syncthreads();

    for (int s = 0; s < T; ++s) {
        const int t = dir ? (T - 1 - s) : s;
        const float* __restrict__ gx = Gx + (size_t)t * G;
        if (s + 1 < T) {
            const int tn = dir ? (t - 1) : (t + 1);
            __builtin_prefetch(Gx + (size_t)tn * G + (tid & (G - 1)), 0, 0);
        }
        const float4* __restrict__ h4 = (const float4*)h;
        for (int j = tid; j < G; j += nth) {
            const float4* __restrict__ wr = (const float4*)(Whh + (size_t)j * Hh);
            float acc = gx[j];
            for (int kq = 0; kq < H4; ++kq) {
                const float4 w = wr[kq];
                const float4 hv = h4[kq];
                acc = fmaf(w.x, hv.x, acc);
                acc = fmaf(w.y, hv.y, acc);
                acc = fmaf(w.z, hv.z, acc);
                acc = fmaf(w.w, hv.w, acc);
            }
            g[j] = acc;
        }
        __syncthreads();
        for (int j = tid; j < Hh; j += nth) {
            const float ig = sigf(g[j]);
            const float fg = sigf(g[Hh + j]);
            const float gg = tanhf(g[2 * Hh + j]);
            const float og = sigf(g[3 * Hh + j]);
            const float cn = fg * c[j] + ig * gg;
            c[j] = cn;
            const float hn = og * tanhf(cn);
            h[j] = hn;
            out[(size_t)t * outStride + (size_t)dir * Hh + j] = hn;
        }
        __syncthreads();
    }
}

// ---------------------------------------------------------------------------
// Concat [note_out(512) | y(11)] -> 2048 x 523 buffer for the contract layer.
// ---------------------------------------------------------------------------
__global__ void concat_ny(const float* __restrict__ note,
                          const float* __restrict__ y,
                          float* __restrict__ o, int T)
{
    const int i = blockIdx.x * blockDim.x + threadIdx.x;
    const int total = T * 523;
    if (i < total) {
        const int t = i / 523, cdx = i % 523;
        o[i] = (cdx < 512) ? note[(size_t)t * 512 + cdx]
                           : y[(size_t)t * 11 + (cdx - 512)];
    }
}

// ---------------------------------------------------------------------------
// Multi-head context attention over one segment of T rows of width D (8 heads).
//   a = tanh(x@W^T + b);  sim[t][h] = a[t,head h]·ctx[h];  w = softmax_t(sim);
//   out[d] = sum_t x[t][d] * w[t][head(d)]
// ---------------------------------------------------------------------------
__global__ void ctx_att(const float* __restrict__ X,
                        const float* __restrict__ W,
                        const float* __restrict__ b,
                        const float* __restrict__ ctx,
                        float* __restrict__ out, int T, int D)
{
    const int seg = blockIdx.x;
    const float* __restrict__ x = X + (size_t)seg * T * D;
    const int hs = D / 8;

    __shared__ float sim[128 * 8];
    __shared__ float wts[128 * 8];

    const int tid = threadIdx.x;
    const int nth = blockDim.x;

    for (int p = tid; p < T * 8; p += nth) {
        const int t = p / 8, hh = p % 8;
        float s = 0.f;
        for (int d = 0; d < hs; ++d) {
            const int col = hh * hs + d;
            const float* __restrict__ wrow = W + (size_t)col * D;
            float av = b[col];
            for (int kq = 0; kq < D; ++kq) av = fmaf(x[(size_t)t * D + kq], wrow[kq], av);
            s = fmaf(tanhf(av), ctx[hh * hs + d], s);
        }
        sim[p] = s;
    }
    __syncthreads();
    if (tid < 8) {
        float mx = -1e30f;
        for (int t = 0; t < T; ++t) mx = fmaxf(mx, sim[t * 8 + tid]);
        float sum = 0.f;
        for (int t = 0; t < T; ++t) {
            const float e = __expf(sim[t * 8 + tid] - mx);
            wts[t * 8 + tid] = e;
            sum += e;
        }
        const float inv = 1.f / sum;
        for (int t = 0; t < T; ++t) wts[t * 8 + tid] *= inv;
    }
    __syncthreads();
    for (int d = tid; d < D; d += nth) {
        const int hh = d / hs;
        float acc = 0.f;
        for (int t = 0; t < T; ++t) acc = fmaf(x[(size_t)t * D + d], wts[t * 8 + hh], acc);
        out[(size_t)seg * D + d] = acc;
    }
}

// ---------------------------------------------------------------------------
// VAE head: mu/logvar matvecs (16x256), reparameterized z, static vector
//   static = [ x[0,0,4], x[0,0,76], x[0,0,77], z(16) ]   (19 floats)
// ---------------------------------------------------------------------------
__global__ void vae_head(const float* __restrict__ style,
                         const float* __restrict__ Wm, const float* __restrict__ bm,
                         const float* __restrict__ Wv, const float* __restrict__ bv,
                         const float* __restrict__ x,
                         float* __restrict__ mu_out, float* __restrict__ lv_out,
                         float* __restrict__ stat)
{
    const int j = threadIdx.x;
    if (j < 16) {
        float m = bm[j], v = bv[j];
        const float* __restrict__ wm = Wm + (size_t)j * 256;
        const float* __restrict__ wv = Wv + (size_t)j * 256;
        for (int kq = 0; kq < 256; ++kq) {
            m = fmaf(style[kq], wm[kq], m);
            v = fmaf(style[kq], wv[kq], v);
        }
        mu_out[j] = m;
        lv_out[j] = v;
        stat[3 + j] = m + EPS16[j] * __expf(0.5f * v);
    }
    if (j == 16) { stat[0] = x[4]; stat[1] = x[76]; stat[2] = x[77]; }
}

// ---------------------------------------------------------------------------
// Autoregressive output LSTM (FH=512, gates=2048).
// Gpre[t] = note_out[t] @ Wih[:, :512]^T + b  (precomputed via WMMA GEMM).
// 1024 threads; each handles 2 gate rows: float4 Whh dot + float2 Wih tail.
// fc_out done by 11 wave-wide groups with shuffle reduction.
// ---------------------------------------------------------------------------
__global__ void out_lstm_seq(const float* __restrict__ Gpre,
                             const float* __restrict__ Wih,
                             const float* __restrict__ Whh,
                             const float* __restrict__ Wf,
                             const float* __restrict__ bf,
                             const float* __restrict__ stat,
                             float* __restrict__ outs, int T)
{
    const int FH = 512, G = 2048;
    __shared__ __align__(16) float h[512];
    __shared__ float c[512];
    __shared__ float g[2048];
    __shared__ float tail[32];

    const int tid = threadIdx.x;
    const int nth = blockDim.x;

    for (int j = tid; j < FH; j += nth) { h[j] = 0.f; c[j] = 0.f; }
    if (tid < 11) tail[tid] = 0.f;               // prev_out starts at 0
    if (tid < 19) tail[11 + tid] = stat[tid];    // static part, constant
    __syncthreads();

    for (int t = 0; t < T; ++t) {
        const float* __restrict__ gx = Gpre + (size_t)t * G;
        if (t + 1 < T) __builtin_prefetch(Gpre + (size_t)(t + 1) * G + tid, 0, 0);
        const float4* __restrict__ h4 = (const float4*)h;
        for (int j = tid; j < G; j += nth) {
            // tail: Wih columns 512..541 = [prev(11) | static(19)], float2-aligned
            const float2* __restrict__ wi = (const float2*)(Wih + (size_t)j * FIN_D + 512);
            const float4* __restrict__ wh = (const float4*)(Whh + (size_t)j * FH);
            float acc = gx[j];
            #pragma unroll
            for (int kq = 0; kq < 15; ++kq) {
                const float2 w = wi[kq];
                acc = fmaf(w.x, tail[2 * kq], acc);
                acc = fmaf(w.y, tail[2 * kq + 1], acc);
            }
            for (int kq = 0; kq < FH / 4; ++kq) {
                const float4 w = wh[kq];
                const float4 hv = h4[kq];
                acc = fmaf(w.x, hv.x, acc);
                acc = fmaf(w.y, hv.y, acc);
                acc = fmaf(w.z, hv.z, acc);
                acc = fmaf(w.w, hv.w, acc);
            }
            g[j] = acc;
        }
        __syncthreads();
        for (int j = tid; j < FH; j += nth) {
            const float ig = sigf(g[j]);
            const float fg = sigf(g[FH + j]);
            const float gg = tanhf(g[2 * FH + j]);
            const float og = sigf(g[3 * FH + j]);
            const float cn = fg * c[j] + ig * gg;
            c[j] = cn;
            h[j] = og * tanhf(cn);
        }
        __syncthreads();
        if (tid < 11 * 32) {                     // 11 full waves, shuffle-reduced
            const int o = tid >> 5, ln = tid & 31;
            const float* __restrict__ wf = Wf + (size_t)o * FH;
            float p = 0.f;
            for (int kq = ln; kq < FH; kq += 32) p = fmaf(wf[kq], h[kq], p);
            #pragma unroll
            for (int off = 16; off > 0; off >>= 1) p += __shfl_down(p, off, 32);
            if (ln == 0) {
                const float vv = p + bf[o];
                tail[o] = vv;
                outs[(size_t)t * 11 + o] = vv;
            }
        }
        __syncthreads();
    }
}

// ---------------------------------------------------------------------------
// Host-side orchestration
// ---------------------------------------------------------------------------
static inline void gemm(const float* A, const float* W, const float* b, float* C,
                        int M, int N, int K, int lda, int ldw, int ldc, int act,
                        hipStream_t s)
{
    dim3 grid((N + 15) / 16, (M + 63) / 64);
    gemm_wmma_f32<<<grid, dim3(32, 4), 0, s>>>(A, W, b, C, M, N, K, lda, ldw, ldc, act);
}

extern "C" void kernel_launch(void* const* d_in, const int* in_sizes, int n_in,
                              void* d_out, int out_size, void* d_ws, size_t ws_size,
                              hipStream_t stream)
{
    (void)in_sizes; (void)n_in; (void)out_size; (void)ws_size;

    const float* x = (const float*)d_in[0];
    const float* y = (const float*)d_in[1];
    // d_in[2] = measure_numbers (regular N/NPM structure — not needed)

    const float* conW  = (const float*)d_in[3];
    const float* conB  = (const float*)d_in[4];
    const float* fc1W  = (const float*)d_in[5];
    const float* fc1B  = (const float*)d_in[6];
    const float* fc2W  = (const float*)d_in[7];
    const float* fc2B  = (const float*)d_in[8];
    const float* fc3W  = (const float*)d_in[9];
    const float* fc3B  = (const float*)d_in[10];
    const float* foW   = (const float*)d_in[11];
    const float* foB   = (const float*)d_in[12];
    const float* faW   = (const float*)d_in[13];
    const float* faB   = (const float*)d_in[14];
    const float* faC   = (const float*)d_in[15];
    const float* lWhh[2][2] = {{(const float*)d_in[16], (const float*)d_in[19]},
                               {(const float*)d_in[22], (const float*)d_in[25]}};
    const float* lWih[2][2] = {{(const float*)d_in[17], (const float*)d_in[20]},
                               {(const float*)d_in[23], (const float*)d_in[26]}};
    const float* lB[2][2]   = {{(const float*)d_in[18], (const float*)d_in[21]},
                               {(const float*)d_in[24], (const float*)d_in[27]}};
    const float* meanW = (const float*)d_in[28];
    const float* meanB = (const float*)d_in[29];
    const float* maW   = (const float*)d_in[30];
    const float* maB   = (const float*)d_in[31];
    const float* maC   = (const float*)d_in[32];
    const float* neWhh[2] = {(const float*)d_in[33], (const float*)d_in[36]};
    const float* neWih[2] = {(const float*)d_in[34], (const float*)d_in[37]};
    const float* neB[2]   = {(const float*)d_in[35], (const float*)d_in[38]};
    const float* olWhh = (const float*)d_in[39];
    const float* olWih = (const float*)d_in[40];
    const float* olB   = (const float*)d_in[41];
    const float* peWhh[2][2] = {{(const float*)d_in[42], (const float*)d_in[45]},
                                {(const float*)d_in[48], (const float*)d_in[51]}};
    const float* peWih[2][2] = {{(const float*)d_in[43], (const float*)d_in[46]},
                                {(const float*)d_in[49], (const float*)d_in[52]}};
    const float* peB[2][2]   = {{(const float*)d_in[44], (const float*)d_in[47]},
                                {(const float*)d_in[50], (const float*)d_in[53]}};
    const float* varW = (const float*)d_in[54];
    const float* varB = (const float*)d_in[55];

    float* out = (float*)d_out;
    float* outs_o   = out;                    // 2048*11
    float* mu_o     = out + 22528;            // 16
    float* lv_o     = out + 22544;            // 16
    float* note_out = out + 22560;            // 2048*512

    // workspace layout (floats)
    float* ws = (float*)d_ws;
    float* h1    = ws;                       // 2048*256
    float* h2    = ws + 524288;              // 2048*256
    float* h3    = ws + 1048576;             // 2048*256
    float* gxf   = ws + 1572864;             // 2048*1024 (reused; Gpre lower half)
    float* gxb   = ws + 3670016;             // 2048*1024 (reused; Gpre upper half)
    float* l0out = ws + 5767168;             // 2048*512
    float* catb  = ws + 6815744;             // 2048*523
    float* pc    = ws + 7886848;             // 2048*128
    float* pne   = ws + 8148992;             // 2048*256
    float* meas  = ws + 8673280;             // 128*256
    float* pe0   = ws + 8706048;             // 128*256
    float* pe1   = ws + 8738816;             // 128*256
    float* style = ws + 8771584;             // 256
    float* statv = ws + 8771840;             // 19
    float* gpre  = gxf;                      // 2048*2048, reuses gxf+gxb region

    // ---- FC stack (WMMA GEMMs, relu) ----
    gemm(x,  fc1W, fc1B, h1, T_N, H_D, IN_D, IN_D, IN_D, H_D, 1, stream);
    gemm(h1, fc2W, fc2B, h2, T_N, H_D, H_D,  H_D,  H_D,  H_D, 1, stream);
    gemm(h2, fc3W, fc3B, h3, T_N, H_D, H_D,  H_D,  H_D,  H_D, 1, stream);

    // ---- biLSTM layer 0 (in=256) ----
    gemm(h3, lWih[0][0], lB[0][0], gxf, T_N, 4*H_D, H_D, H_D, H_D, 4*H_D, 0, stream);
    gemm(h3, lWih[0][1], lB[0][1], gxb, T_N, 4*H_D, H_D, H_D, H_D, 4*H_D, 0, stream);
    bilstm_seq<<<2, 1024, 0, stream>>>(gxf, lWhh[0][0], gxb, lWhh[0][1],
                                       l0out, T_N, H_D, 2*H_D);

    // ---- biLSTM layer 1 (in=512) -> note_out (in d_out) ----
    gemm(l0out, lWih[1][0], lB[1][0], gxf, T_N, 4*H_D, 2*H_D, 2*H_D, 2*H_D, 4*H_D, 0, stream);
    gemm(l0out, lWih[1][1], lB[1][1], gxb, T_N, 4*H_D, 2*H_D, 2*H_D, 2*H_D, 4*H_D, 0, stream);
    bilstm_seq<<<2, 1024, 0, stream>>>(gxf, lWhh[1][0], gxb, lWhh[1][1],
                                       note_out, T_N, H_D, 2*H_D);

    // ---- contract: relu(concat(note_out, y) @ Wc^T + bc) ----
    concat_ny<<<(T_N * 523 + 255) / 256, 256, 0, stream>>>(note_out, y, catb, T_N);
    gemm(catb, conW, conB, pc, T_N, E_D, 523, 523, 523, E_D, 1, stream);

    // ---- note encoder biLSTM (E=128) -> pne ----
    gemm(pc, neWih[0], neB[0], gxf, T_N, 4*E_D, E_D, E_D, E_D, 4*E_D, 0, stream);
    gemm(pc, neWih[1], neB[1], gxb, T_N, 4*E_D, E_D, E_D, E_D, 4*E_D, 0, stream);
    bilstm_seq<<<2, 1024, 0, stream>>>(gxf, neWhh[0], gxb, neWhh[1],
                                       pne, T_N, E_D, 2*E_D);

    // ---- per-measure attention: 128 segments of 16 notes ----
    ctx_att<<<T_N / NPM_D, 256, 0, stream>>>(pne, maW, maB, maC, meas, NPM_D, 2*E_D);

    // ---- perf encoder biLSTM x2 over 128 measures ----
    const int M_ = T_N / NPM_D;
    gemm(meas, peWih[0][0], peB[0][0], gxf, M_, 4*E_D, 2*E_D, 2*E_D, 2*E_D, 4*E_D, 0, stream);
    gemm(meas, peWih[0][1], peB[0][1], gxb, M_, 4*E_D, 2*E_D, 2*E_D, 2*E_D, 4*E_D, 0, stream);
    bilstm_seq<<<2, 1024, 0, stream>>>(gxf, peWhh[0][0], gxb, peWhh[0][1],
                                       pe0, M_, E_D, 2*E_D);
    gemm(pe0, peWih[1][0], peB[1][0], gxf, M_, 4*E_D, 2*E_D, 2*E_D, 2*E_D, 4*E_D, 0, stream);
    gemm(pe0, peWih[1][1], peB[1][1], gxb, M_, 4*E_D, 2*E_D, 2*E_D, 2*E_D, 4*E_D, 0, stream);
    bilstm_seq<<<2, 1024, 0, stream>>>(gxf, peWhh[1][0], gxb, peWhh[1][1],
                                       pe1, M_, E_D, 2*E_D);

    // ---- final attention over 128 measures -> style ----
    ctx_att<<<1, 256, 0, stream>>>(pe1, faW, faB, faC, style, M_, 2*E_D);

    // ---- VAE head: mu, logvar, z, static vector ----
    vae_head<<<1, 32, 0, stream>>>(style, meanW, meanB, varW, varB, x,
                                   mu_o, lv_o, statv);

    // ---- output LSTM: big WMMA pre-projection then sequential recurrence ----
    // Gpre = note_out @ Wih[:, :512]^T + b   (M=2048, N=2048, K=512)
    gemm(note_out, olWih, olB, gpre, T_N, 4*FH_D, FH_D, FH_D, FIN_D, 4*FH_D, 0, stream);
    out_lstm_seq<<<1, 1024, 0, stream>>>(gpre, olWih, olWhh, foW, foB, statv,
                                         outs_o, T_N);
}